// BehaviorSnake_30296699306446
// MI455X (gfx1250) — compile-verified
//
#include <hip/hip_runtime.h>
#include <stdint.h>

// BehaviorSnake CA, fused single-pass stencil for gfx1250 (MI455X).
// (B,C,H,W) = (8,10,512,512) f32. Channels: 0..3 elem one-hot (EMPTY,WALL,ACID,SNAKE),
// 6 = direction, 7 = energy. Toroidal shifts. Bandwidth-bound: one HBM pass.
// gfx1250 paths used: LDS ping-pong CA (ds ops + split barriers), async
// global->LDS staging of final-phase data (global_load_async_to_lds_b32 /
// s_wait_asynccnt) overlapped with the 4 dependent CA iterations.

constexpr int Bc = 8, Cc = 10, Hc = 512, Wc = 512;
constexpr int TILE = 32;
constexpr int R    = 5;               // halo: 4 CA iterations + 1 for final in_dir neighbor
constexpr int RX   = TILE + 2 * R;    // 42
constexpr int RN   = RX * RX;         // 1764
constexpr int TN   = TILE * TILE;     // 1024

// State word bits (per cell, LDS-resident across the 4 dependent iterations):
//  [0]=E [1]=W [2]=A [3]=S   elem channel == 1.0f flags (mutated by EMPTY writes)
//  [4]=emptied  [5]=does_become_trail  [6]=does_become_snake  [7]=does_turn
//  [9:8]=dir_came   [10]=postS [11]=postW  (post-loop SNAKE/WALL==1, for in_dir)
//  [12]=dir valid (dir in {0,1,2,3} exactly)  [14:13]=dir code  [15]=energy > 0

__device__ __forceinline__ unsigned lds_off(const void* p) {
    // raw LDS byte offset (async-to-LDS VDST operand wants the DS address)
    return (unsigned)(uintptr_t)(__attribute__((address_space(3))) const void*)p;
}

__device__ __forceinline__ void async_ld_b32(unsigned lds_byte, unsigned gbyte_off,
                                             const float* sbase) {
    asm volatile("global_load_async_to_lds_b32 %0, %1, %2"
                 :: "v"(lds_byte), "v"(gbyte_off), "s"(sbase)
                 : "memory");
}

__global__ __launch_bounds__(256)
void snake_step_kernel(const float* __restrict__ world,
                       const float* __restrict__ rmov,
                       const float* __restrict__ relem,
                       float* __restrict__ out)
{
    __shared__ unsigned stA[RN];
    __shared__ unsigned stB[RN];
    __shared__ float    gbuf[11][TN];   // planes 0..9 = world ch0..9, 10 = rand_element

    const int tid = threadIdx.x;
    const int b   = blockIdx.z;
    const int h0  = blockIdx.y * TILE;
    const int w0  = blockIdx.x * TILE;

    const size_t planeHW = (size_t)Hc * Wc;
    const float* wB  = world + (size_t)b * Cc * planeHW;
    const float* rmB = rmov  + (size_t)b * planeHW;
    const float* reB = relem + (size_t)b * planeHW;
    float*       oB  = out   + (size_t)b * Cc * planeHW;

    // ---- Kick off async global->LDS staging of final-phase data.
    // Each thread stages exactly the pixels it will consume later, so a
    // per-wave s_wait_asynccnt 0 (no barrier) is sufficient before use.
    for (int t = tid; t < TN; t += 256) {
        const int th = t >> 5, tw = t & 31;
        const unsigned pixb = (unsigned)((h0 + th) * Wc + (w0 + tw)) * 4u;
#pragma unroll
        for (int c = 0; c < 10; ++c)
            async_ld_b32(lds_off(&gbuf[c][t]), pixb, wB + (size_t)c * planeHW);
        async_ld_b32(lds_off(&gbuf[10][t]), pixb, reB);
    }

    // ---- Stage 0: build packed state over the 42x42 region (toroidal wrap) ----
    for (int r = tid; r < RN; r += 256) {
        const int rh = r / RX, rw = r % RX;
        const int gh = (h0 - R + rh) & (Hc - 1);
        const int gw = (w0 - R + rw) & (Wc - 1);
        const size_t pix = (size_t)gh * Wc + gw;
        const float e0 = wB[0 * planeHW + pix];
        const float e1 = wB[1 * planeHW + pix];
        const float e2 = wB[2 * planeHW + pix];
        const float e3 = wB[3 * planeHW + pix];
        const float dv = wB[6 * planeHW + pix];
        const float en = wB[7 * planeHW + pix];
        const float rm = rmB[pix];
        unsigned s = 0;
        if (e0 == 1.0f) s |= 1u;
        if (e1 == 1.0f) s |= 2u;
        if (e2 == 1.0f) s |= 4u;
        if (e3 == 1.0f) s |= 8u;
        if (rm < 0.1f)  s |= 1u << 7;            // does_turn seed
        if      (dv == 0.0f) s |= (1u << 12);
        else if (dv == 1.0f) s |= (1u << 12) | (1u << 13);
        else if (dv == 2.0f) s |= (1u << 12) | (2u << 13);
        else if (dv == 3.0f) s |= (1u << 12) | (3u << 13);
        if (en > 0.0f) s |= 1u << 15;
        stA[r] = s;
    }
    __syncthreads();

    // Movement deltas per angle: 0:+w  1:+h  2:-w  3:-h
    const int dh[4] = {0, 1, 0, -1};
    const int dw[4] = {1, 0, -1, 0};

    // ---- Stages 1..4: the four dependent CA iterations, LDS ping-pong ----
    // (async staging above overlaps with this whole phase)
#pragma unroll
    for (int a = 0; a < 4; ++a) {
        unsigned* __restrict__ in = (a & 1) ? stB : stA;
        unsigned* __restrict__ o  = (a & 1) ? stA : stB;
        const int m = a + 1;                     // validity margin shrinks by 1/iter
        for (int r = tid; r < RN; r += 256) {
            const int rh = r / RX, rw = r % RX;
            if (rh < m || rh >= RX - m || rw < m || rw >= RX - m) continue;
            const unsigned s  = in[r];
            const unsigned sm = in[(rh - dh[a]) * RX + (rw - dw[a])];  // snake source
            const unsigned sp = in[(rh + dh[a]) * RX + (rw + dw[a])];  // blocked probe
            const bool isa_m = (sm & 8u) && (sm & (1u << 12)) &&
                               (((sm >> 13) & 3u) == (unsigned)a);
            const bool dbta  = isa_m && !(s & 2u);                     // snake arrives
            const bool blk   = (sp & 2u) || (sp & 8u);                 // wall|snake ahead
            const bool isa_x = (s & 8u) && (s & (1u << 12)) &&
                               (((s >> 13) & 3u) == (unsigned)a);
            unsigned ns = s;
            if (isa_x)        ns |= 1u << 5;                           // trail
            if (dbta && blk)  ns |= 1u << 7;                           // turn
            if (dbta) {
                if (s & (1u << 6)) {                                   // second arrival:
                    ns = (ns & ~0xFu) | 1u;                            //  EMPTY one-hot
                    ns |= 1u << 4;                                     //  emptied
                    ns = (ns & ~(7u << 12)) | (1u << 12);              //  dir -> 0.0
                    ns &= ~(1u << 15);                                 //  energy -> 0
                }
                ns |= 1u << 6;                                         // does_become_snake
                ns = (ns & ~(3u << 8)) | ((unsigned)a << 8);           // dir_came = a
            }
            if (a == 3) {
                // fold in post-loop SNAKE/WALL ==1 flags (for in_dir checks)
                const bool fdbs = (ns & (1u << 6)) != 0;
                const bool fdbt = (ns & (1u << 5)) != 0;
                const bool enp  = (ns & (1u << 15)) != 0;
                const bool postS = fdbs || (fdbt && !enp) ||
                                   (!fdbs && !fdbt && (ns & 8u));
                const bool postW = !fdbs && !fdbt && (ns & 2u);
                if (postS) ns |= 1u << 10;
                if (postW) ns |= 1u << 11;
            }
            o[r] = ns;
        }
        __syncthreads();
    }

    // All of this wave's async staging must have landed in LDS before reading.
    asm volatile("s_wait_asynccnt 0x0" ::: "memory");

    // ---- Final stage: turn resolution + write all 10 output channels ----
    for (int t = tid; t < TN; t += 256) {
        const int th = t >> 5, tw = t & 31;
        const int rh = R + th, rw = R + tw;
        const size_t pix = (size_t)(h0 + th) * Wc + (w0 + tw);

        const unsigned s = stA[rh * RX + rw];          // final state (after iter 3)
        const bool dbt   = (s & (1u << 5)) != 0;
        const bool dbs   = (s & (1u << 6)) != 0;
        const bool turn0 = (s & (1u << 7)) != 0;
        const unsigned dc = (s >> 8) & 3u;

        const float re   = gbuf[10][t];
        const float dirv = gbuf[6][t];
        const float env  = gbuf[7][t];
        const bool  ws   = (gbuf[3][t] == 1.0f);       // was_snake (original)

        // turned = (dir_came + 1 - 2*(re<0.5)) mod 4  ==  (dc + 1 + 2*t) & 3
        const unsigned tf = (re < 0.5f) ? 1u : 0u;
        const unsigned turned = (dc + 1u + 2u * tf) & 3u;
        const int ndh = (turned == 1u) ? 1 : ((turned == 3u) ? -1 : 0);
        const int ndw = (turned == 0u) ? 1 : ((turned == 2u) ? -1 : 0);
        const unsigned sy = stA[(rh + ndh) * RX + (rw + ndw)]; // post-loop neighbor
        const bool inS = (sy & (1u << 10)) != 0;
        const bool inW = (sy & (1u << 11)) != 0;
        const bool turn = turn0 && !inS && !inW;

        const bool postS = (s & (1u << 10)) != 0;
        const unsigned dcf = turn ? turned : dc;
        const float new_dir = (postS && !ws) ? (float)dcf : dirv;
        const float w6p = (dbs || dbt) ? 0.0f : dirv;
        const float chDir = postS ? new_dir : w6p;
        const float w7p = (dbs || dbt) ? 0.0f : env;
        const float chEn = postS ? (env - 0.1f) : w7p;

        float o0, o1, o2, o3, o4, o5, o8, o9;
        if (dbs) {                                    // snake moves in (wins over trail)
            o0 = o1 = o2 = o4 = o5 = o8 = o9 = 0.0f; o3 = 1.0f;
        } else if (dbt) {                             // trail: acid/empty/snake
            o0 = o1 = o2 = o3 = o4 = o5 = o8 = o9 = 0.0f;
            if (env > 0.0f) { if (re < 0.05f) o0 = 1.0f; else o2 = 1.0f; }
            else o3 = 1.0f;
        } else {                                      // untouched: pass-through
            o0 = gbuf[0][t]; o1 = gbuf[1][t];
            o2 = gbuf[2][t]; o3 = gbuf[3][t];
            o4 = gbuf[4][t]; o5 = gbuf[5][t];
            o8 = gbuf[8][t]; o9 = gbuf[9][t];
        }
        oB[0 * planeHW + pix] = o0;  oB[1 * planeHW + pix] = o1;
        oB[2 * planeHW + pix] = o2;  oB[3 * planeHW + pix] = o3;
        oB[4 * planeHW + pix] = o4;  oB[5 * planeHW + pix] = o5;
        oB[6 * planeHW + pix] = chDir;
        oB[7 * planeHW + pix] = chEn;
        oB[8 * planeHW + pix] = o8;  oB[9 * planeHW + pix] = o9;
    }
}

extern "C" void kernel_launch(void* const* d_in, const int* in_sizes, int n_in,
                              void* d_out, int out_size, void* d_ws, size_t ws_size,
                              hipStream_t stream) {
    (void)in_sizes; (void)n_in; (void)out_size; (void)d_ws; (void)ws_size;
    const float* world = (const float*)d_in[0];
    const float* rmov  = (const float*)d_in[1];
    // d_in[2] = rand_interact: unused by the reference computation
    const float* relem = (const float*)d_in[3];
    float* out = (float*)d_out;

    dim3 grid(Wc / TILE, Hc / TILE, Bc);   // 16 x 16 x 8 tiles
    dim3 block(256);                       // 8 wave32 waves
    snake_step_kernel<<<grid, block, 0, stream>>>(world, rmov, relem, out);
}